// QGCN_25494925869657
// MI455X (gfx1250) — compile-verified
//
#include <hip/hip_runtime.h>
#include <math.h>

typedef __attribute__((ext_vector_type(2))) float v2f;
typedef __attribute__((ext_vector_type(8))) float v8f;
typedef __attribute__((ext_vector_type(4))) unsigned int u32x4;
typedef __attribute__((ext_vector_type(8))) unsigned int u32x8;

#define HDIM 128
#define INDIM 512
#define ODIM 7
#define KCHUNK 64   // K-rows of B staged per TDM transfer (64*128*4B = 32 KB)

// ---------------------------------------------------------------------------
// degree / norm kernels
// ---------------------------------------------------------------------------
__global__ void k_fill1(float* __restrict__ p, int n) {
    int i = blockIdx.x * blockDim.x + threadIdx.x;
    if (i < n) p[i] = 1.0f;  // self loop contributes 1 to every degree
}

__global__ void k_deg(const int* __restrict__ dst, float* __restrict__ deg, int E) {
    int e = blockIdx.x * blockDim.x + threadIdx.x;
    if (e < E) atomicAdd(&deg[dst[e]], 1.0f);
}

__global__ void k_rsqrt(float* __restrict__ p, int n) {
    int i = blockIdx.x * blockDim.x + threadIdx.x;
    if (i < n) p[i] = rsqrtf(p[i]);
}

// ---------------------------------------------------------------------------
// TDM: DMA a contiguous run of `nelem` f32 (nelem <= 65535) from global to LDS.
// Builds the D# (groups 0+1; groups 2/3 NULL -> <=2D tensor) in SGPR tuples
// and issues TENSOR_LOAD_TO_LDS. Caller must s_wait_tensorcnt + barrier.
// ---------------------------------------------------------------------------
__device__ __forceinline__ void tdm_copy_f32(const float* gsrc, float* lds_dst,
                                             unsigned nelem) {
    unsigned long long ga = (unsigned long long)(uintptr_t)gsrc;
    unsigned lds = (unsigned)(uintptr_t)lds_dst;  // addr[31:0] = LDS byte offset

    u32x4 g0;
    g0.x = 1u;                                          // count=1, user D#
    g0.y = lds;                                         // lds_addr (bytes)
    g0.z = (unsigned)(ga & 0xffffffffu);                // global_addr[31:0]
    g0.w = (unsigned)((ga >> 32) & 0x01ffffffu)         // global_addr[56:32]
           | (2u << 30);                                // type = 2 ("image")

    u32x8 g1;
    g1.s0 = (2u << 16);                  // wg_mask=0, data_size=2 (4 bytes)
    g1.s1 = (nelem & 0xffffu) << 16;     // tensor_dim0[15:0]  (bits 63:48)
    g1.s2 = ((nelem >> 16) & 0xffffu)    // tensor_dim0[31:16] (bits 79:64)
            | (1u << 16);                // tensor_dim1 = 1    (bits 111:80)
    g1.s3 = (nelem & 0xffffu) << 16;     // tile_dim0 (bits 127:112); dim1 hi = 0
    g1.s4 = 0u;                          // tile_dim1 = 0 (1-D tile), tile_dim2 = 0
    g1.s5 = nelem;                       // tensor_dim0_stride[31:0]
    g1.s6 = 0u;                          // stride hi bits
    g1.s7 = 0u;

    asm volatile("tensor_load_to_lds %0, %1" :: "s"(g0), "s"(g1) : "memory");
}

// ---------------------------------------------------------------------------
// WMMA GEMM: C[M,N] = A[M,K] @ B[K,N], N == 128, M % 16 == 0, K % 64 == 0.
// One block = 8 waves; wave w computes the 16x16 tile at (blockIdx.x*16, w*16).
// B is staged chunk-wise into LDS by wave 0 via the Tensor Data Mover; all
// waves then feed V_WMMA_F32_16X16X4_F32 from LDS (ds_load) while A streams
// from global (b64 loads). No divergence inside a wave: EXEC all-1s for WMMA.
// ---------------------------------------------------------------------------
__global__ void k_gemm_wmma(const float* __restrict__ A, const float* __restrict__ B,
                            float* __restrict__ C, int K) {
    __shared__ float sB[KCHUNK * HDIM];  // 32 KB

    const int N = HDIM;
    int lane = threadIdx.x & 31;
    int wave = threadIdx.x >> 5;
    int half = lane >> 4;        // 0: lanes 0-15, 1: lanes 16-31
    int l    = lane & 15;
    int m0 = blockIdx.x * 16;
    int n0 = wave * 16;
    int h2 = half * 2;

    // A 16x4 layout: lanes<16 hold K0(v0),K1(v1); lanes>=16 hold K2,K3
    const float* arow = A + (size_t)(m0 + l) * K + h2;

    v8f acc = {};
    for (int kc = 0; kc < K; kc += KCHUNK) {
        __syncthreads();                       // sB safe to overwrite
        if (threadIdx.x < 32) {                // wave 0 drives the DMA
            tdm_copy_f32(B + (size_t)kc * N, sB, KCHUNK * N);
            __builtin_amdgcn_s_wait_tensorcnt(0);
        }
        __syncthreads();                       // sB ready for all waves

#pragma unroll
        for (int kk = 0; kk < KCHUNK; kk += 4) {
            v2f a = *(const v2f*)(arow + kc + kk);
            // B 4x16 layout mirrored over lanes: VGPR0 = rows K0(l<16)/K2(l>=16)
            v2f b;
            b.x = sB[(kk + h2) * N + n0 + l];
            b.y = sB[(kk + h2 + 1) * N + n0 + l];
            acc = __builtin_amdgcn_wmma_f32_16x16x4_f32(
                /*neg_a=*/false, a, /*neg_b=*/false, b,
                /*c_mod=*/(short)0, acc, /*reuse_a=*/false, /*reuse_b=*/false);
        }
    }

    // C/D layout: VGPR g -> row m0 + g + half*8, col n0 + l
    float* crow = C + (size_t)(m0 + half * 8) * N + n0 + l;
#pragma unroll
    for (int g = 0; g < 8; ++g) crow[(size_t)g * N] = acc[g];
}

// ---------------------------------------------------------------------------
// agg[i,:] = H[i,:] * dinv[i]^2   (self-loop term of the scatter sum)
// ---------------------------------------------------------------------------
__global__ void k_selfinit(const float* __restrict__ H, const float* __restrict__ dinv,
                           float* __restrict__ agg, int total) {
    int t = blockIdx.x * blockDim.x + threadIdx.x;
    if (t >= total) return;
    float di = dinv[t >> 7];
    agg[t] = H[t] * di * di;
}

// ---------------------------------------------------------------------------
// edge scatter: one wave per edge, 4 feats per lane (float4 gather,
// coalesced global_atomic_add_f32 scatter).
// ---------------------------------------------------------------------------
__global__ void k_scatter(const int* __restrict__ src, const int* __restrict__ dst,
                          const float* __restrict__ dinv, const float* __restrict__ H,
                          float* __restrict__ agg, int E) {
    int t = blockIdx.x * blockDim.x + threadIdx.x;
    int e = t >> 5;
    int lane = t & 31;
    if (e >= E) return;
    int s = src[e], d = dst[e];
    float norm = dinv[s] * dinv[d];
    float4 v = ((const float4*)(H + (size_t)s * HDIM))[lane];
    float* ad = agg + (size_t)d * HDIM + lane * 4;
    atomicAdd(ad + 0, v.x * norm);
    atomicAdd(ad + 1, v.y * norm);
    atomicAdd(ad + 2, v.z * norm);
    atomicAdd(ad + 3, v.w * norm);
}

// ---------------------------------------------------------------------------
// out[i,f] = tanh(agg[i,f] + b[f])
// ---------------------------------------------------------------------------
__global__ void k_bias_tanh(const float* __restrict__ agg, const float* __restrict__ b,
                            float* __restrict__ out, int total) {
    int t = blockIdx.x * blockDim.x + threadIdx.x;
    if (t >= total) return;
    out[t] = tanhf(agg[t] + b[t & (HDIM - 1)]);
}

// ---------------------------------------------------------------------------
// head: logits = H @ Wl + bl ; logp = logits - logsumexp(logits)
// ---------------------------------------------------------------------------
__global__ void k_head(const float* __restrict__ H, const float* __restrict__ Wl,
                       const float* __restrict__ bl, float* __restrict__ out, int Nn) {
    int i = blockIdx.x * blockDim.x + threadIdx.x;
    if (i >= Nn) return;
    const float* h = H + (size_t)i * HDIM;
    float logit[ODIM];
#pragma unroll
    for (int o = 0; o < ODIM; ++o) logit[o] = bl[o];
    for (int f = 0; f < HDIM; ++f) {
        float hv = h[f];
#pragma unroll
        for (int o = 0; o < ODIM; ++o) logit[o] = fmaf(hv, Wl[f * ODIM + o], logit[o]);
    }
    float mx = logit[0];
#pragma unroll
    for (int o = 1; o < ODIM; ++o) mx = fmaxf(mx, logit[o]);
    float sum = 0.0f;
#pragma unroll
    for (int o = 0; o < ODIM; ++o) sum += expf(logit[o] - mx);
    float lse = mx + logf(sum);
    float* lg = out + (size_t)i * ODIM;
    float* lp = out + (size_t)Nn * ODIM + (size_t)i * ODIM;
#pragma unroll
    for (int o = 0; o < ODIM; ++o) { lg[o] = logit[o]; lp[o] = logit[o] - lse; }
}

// ---------------------------------------------------------------------------
extern "C" void kernel_launch(void* const* d_in, const int* in_sizes, int n_in,
                              void* d_out, int out_size, void* d_ws, size_t ws_size,
                              hipStream_t stream) {
    const float* x  = (const float*)d_in[0];
    const int*   ei = (const int*)d_in[1];
    const float* W1 = (const float*)d_in[2];
    const float* b1 = (const float*)d_in[3];
    const float* W2 = (const float*)d_in[4];
    const float* b2 = (const float*)d_in[5];
    const float* Wl = (const float*)d_in[6];
    const float* bl = (const float*)d_in[7];
    float* out = (float*)d_out;

    const int Nn = in_sizes[0] / INDIM;     // 100000
    const int E  = in_sizes[1] / 2;         // 1600000
    const int* src = ei;
    const int* dst = ei + E;

    // workspace layout: dinv | bufA | bufB
    float* ws   = (float*)d_ws;
    float* dinv = ws;
    float* bufA = ws + (((size_t)Nn + 63) & ~(size_t)63);
    float* bufB = bufA + (size_t)Nn * HDIM;

    const int T = 256;
    const int totalF = Nn * HDIM;
    dim3 gN((Nn + T - 1) / T);
    dim3 gE((E + T - 1) / T);
    dim3 gE32(((size_t)E * 32 + T - 1) / T);
    dim3 gF((totalF + T - 1) / T);
    dim3 gGemm(Nn / 16);   // 6250 blocks, 8 waves each -> full N=128 coverage

    // --- degrees / normalization ---
    k_fill1<<<gN, T, 0, stream>>>(dinv, Nn);
    k_deg<<<gE, T, 0, stream>>>(dst, dinv, E);
    k_rsqrt<<<gN, T, 0, stream>>>(dinv, Nn);

    // --- layer 1: H1 = x @ W1 ; aggregate ; tanh(+b1) ---
    k_gemm_wmma<<<gGemm, T, 0, stream>>>(x, W1, bufA, INDIM);
    k_selfinit<<<gF, T, 0, stream>>>(bufA, dinv, bufB, totalF);
    k_scatter<<<gE32, T, 0, stream>>>(src, dst, dinv, bufA, bufB, E);
    k_bias_tanh<<<gF, T, 0, stream>>>(bufB, b1, bufA, totalF);

    // --- layer 2: H2 = out1 @ W2 ; aggregate ; tanh(+b2) ---
    k_gemm_wmma<<<gGemm, T, 0, stream>>>(bufA, W2, bufB, HDIM);
    k_selfinit<<<gF, T, 0, stream>>>(bufB, dinv, bufA, totalF);
    k_scatter<<<gE32, T, 0, stream>>>(src, dst, dinv, bufB, bufA, E);
    k_bias_tanh<<<gF, T, 0, stream>>>(bufA, b2, bufB, totalF);

    // --- head: logits + log_softmax ---
    k_head<<<gN, T, 0, stream>>>(bufB, Wl, bl, out, Nn);
}